// OrdinalRegression_59450937311715
// MI455X (gfx1250) — compile-verified
//
#include <hip/hip_runtime.h>

// Problem geometry (fixed by the reference).
#define HW_      (512 * 512)          // pixels per image
#define NCLS     12                   // classes
#define NPIX     (16 * HW_)           // total pixels = 4,194,304
#define NQUADS   (NPIX / 4)           // 1,048,576 pixel-quads
#define NBLOCKS  2048
#define NTHREADS 256                  // 8 wave32 per block; grid*block = 524,288
                                      // -> exactly 2 quads per thread, no divergence

typedef float v2f __attribute__((ext_vector_type(2)));
typedef float v8f __attribute__((ext_vector_type(8)));

// Pass 1: streaming squared-error reduction.
// Each thread: 4 consecutive pixels (int4 target load) x 12 channels (float4 pred
// loads, each channel stream coalesced across the wave -> global_load_b128).
__global__ __launch_bounds__(NTHREADS) void ord_reduce_kernel(
    const float* __restrict__ preds,
    const int*   __restrict__ tgts,
    double*      __restrict__ partials) {
  const int tid   = threadIdx.x;
  const int gtid  = blockIdx.x * NTHREADS + tid;
  const int gsize = gridDim.x * NTHREADS;

  float acc = 0.0f;
  for (int q = gtid; q < NQUADS; q += gsize) {
    const int p0 = q << 2;                 // first pixel of this quad
    const int b  = p0 >> 18;               // / (512*512); quad never crosses images
    const int hw = p0 & (HW_ - 1);
    const int4 t = *reinterpret_cast<const int4*>(tgts + p0);
    const float* pbase = preds + (size_t)b * (NCLS * HW_) + hw;
#pragma unroll
    for (int c = 0; c < NCLS; ++c) {
      const float4 pv = *reinterpret_cast<const float4*>(pbase + (size_t)c * HW_);
      // cumulative one-hot mask: 1 iff c <= target
      const float m0 = (c <= t.x) ? 1.0f : 0.0f;
      const float m1 = (c <= t.y) ? 1.0f : 0.0f;
      const float m2 = (c <= t.z) ? 1.0f : 0.0f;
      const float m3 = (c <= t.w) ? 1.0f : 0.0f;
      const float d0 = pv.x - m0, d1 = pv.y - m1;
      const float d2 = pv.z - m2, d3 = pv.w - m3;
      acc = fmaf(d0, d0, acc);
      acc = fmaf(d1, d1, acc);
      acc = fmaf(d2, d2, acc);
      acc = fmaf(d3, d3, acc);
    }
  }

  // Wave32 reduction via V_WMMA_F32_16X16X4_F32:
  //   A (16x4): VGPR0 = acc, VGPR1 = 0  =>  A[i][0]=acc(i), A[i][2]=acc(i+16)
  //   B (4x16): all ones
  //   D[i][j] = acc(i) + acc(i+16); summing a lane's 8 D regs gives the
  //   half-wave sum (lanes 0-15: lanes {0-7,16-23}; lanes 16-31: {8-15,24-31}).
  v2f a;  a.x = acc;  a.y = 0.0f;
  v2f bo; bo.x = 1.0f; bo.y = 1.0f;
  v8f cz = {};
  v8f d = __builtin_amdgcn_wmma_f32_16x16x4_f32(
      /*neg_a=*/false, a, /*neg_b=*/false, bo,
      /*c_mod=*/(short)0, cz, /*reuse_a=*/false, /*reuse_b=*/false);
  float half = d[0] + d[1] + d[2] + d[3] + d[4] + d[5] + d[6] + d[7];
  float other = __shfl_xor(half, 16, 32);   // combine the two half-wave sums
  const float wavesum = half + other;       // identical on all 32 lanes

  __shared__ double wsum[NTHREADS / 32];
  const int wave = tid >> 5;
  if ((tid & 31) == 0) wsum[wave] = (double)wavesum;
  __syncthreads();
  if (tid == 0) {
    double s = 0.0;                          // fixed-order: deterministic
    for (int w = 0; w < NTHREADS / 32; ++w) s += wsum[w];
    partials[blockIdx.x] = s;
  }
}

// Pass 2: deterministic fixed-order sum of the 2048 double partials + mean.
__global__ __launch_bounds__(NTHREADS) void ord_finalize_kernel(
    const double* __restrict__ partials, float* __restrict__ out) {
  __shared__ double sh[NTHREADS];
  double s = 0.0;
  for (int i = threadIdx.x; i < NBLOCKS; i += NTHREADS) s += partials[i];
  sh[threadIdx.x] = s;
  __syncthreads();
  for (int st = NTHREADS / 2; st > 0; st >>= 1) {
    if ((int)threadIdx.x < st) sh[threadIdx.x] += sh[threadIdx.x + st];
    __syncthreads();
  }
  if (threadIdx.x == 0)
    out[0] = (float)(sh[0] / (double)((size_t)NPIX * (size_t)NCLS));
}

extern "C" void kernel_launch(void* const* d_in, const int* in_sizes, int n_in,
                              void* d_out, int out_size, void* d_ws, size_t ws_size,
                              hipStream_t stream) {
  const float* preds = (const float*)d_in[0];   // [16,12,512,512] f32
  const int*   tgts  = (const int*)d_in[1];     // [16,512,512] integer
  float* out = (float*)d_out;                   // scalar f32 mean
  double* partials = (double*)d_ws;             // 2048 doubles (16 KB)

  ord_reduce_kernel<<<NBLOCKS, NTHREADS, 0, stream>>>(preds, tgts, partials);
  ord_finalize_kernel<<<1, NTHREADS, 0, stream>>>(partials, out);
}